// SimpleGNN_28269474742292
// MI455X (gfx1250) — compile-verified
//
#include <hip/hip_runtime.h>

typedef __attribute__((ext_vector_type(2))) float v2f;
typedef __attribute__((ext_vector_type(8))) float v8f;

#define F_IN  64
#define F_HID 64
#define F_OUT 16

// ---------------- degree / normalization ----------------

__global__ void k_init_deg(float* __restrict__ deg, int N) {
  int i = blockIdx.x * blockDim.x + threadIdx.x;
  if (i < N) deg[i] = 1.0f;               // self loop contributes 1
}

__global__ void k_count_deg(const int* __restrict__ row, float* __restrict__ deg, int E) {
  int e = blockIdx.x * blockDim.x + threadIdx.x;
  if (e < E) atomicAdd(&deg[row[e]], 1.0f);
}

__global__ void k_rsqrt_inplace(float* __restrict__ d, int N) {
  int i = blockIdx.x * blockDim.x + threadIdx.x;
  if (i < N) d[i] = rsqrtf(d[i]);
}

// ---------------- aggregation (self-loop init + edge scatter) ----------------

// aggr[i][f] = dis[i]^2 * x[i][f]   (self-loop term; also zero-initializes)
__global__ void k_init_aggr(const float* __restrict__ x, const float* __restrict__ dis,
                            float* __restrict__ aggr, long long total) {
  long long t = (long long)blockIdx.x * blockDim.x + threadIdx.x;  // N*64 threads
  if (t < total) {
    int i = (int)(t >> 6);
    float w = dis[i] * dis[i];
    aggr[t] = w * x[t];
  }
}

// 16 threads per edge, float4 per thread: 256B coalesced gather + 4 fp32 atomics
__global__ void k_scatter(const float* __restrict__ x,
                          const int* __restrict__ row, const int* __restrict__ col,
                          const float* __restrict__ dis,
                          float* __restrict__ aggr, int E) {
  long long t = (long long)blockIdx.x * blockDim.x + threadIdx.x;
  long long e = t >> 4;
  if (e >= E) return;
  int fg = (int)(t & 15);                 // feature group: 4 floats
  int r = row[e], c = col[e];
  float w = dis[r] * dis[c];
  const float4 xv = *(const float4*)(x + (size_t)r * 64 + fg * 4);
  float* dst = aggr + (size_t)c * 64 + fg * 4;
  atomicAdd(dst + 0, w * xv.x);
  atomicAdd(dst + 1, w * xv.y);
  atomicAdd(dst + 2, w * xv.z);
  atomicAdd(dst + 3, w * xv.w);
}

// ---------------- layer 1 GEMM: h = relu(aggr @ W1 + b1), fp32 WMMA ----------------
// block = 4 waves; wave w computes the 16x16 tile (rows strip, cols 16w..16w+15)

__global__ __launch_bounds__(128) void k_gemm64_relu_wmma(
    const float* __restrict__ A,      // [N,64]
    const float* __restrict__ W,      // [64,64] row-major
    const float* __restrict__ bias,   // [64]
    float* __restrict__ out,          // [N,64]
    int N) {
  int row0 = blockIdx.x * 16;
  if (row0 >= N) return;
  int wave = threadIdx.x >> 5;
  int lane = threadIdx.x & 31;
  int hf   = lane >> 4;                  // half-wave selects K pair
  int l    = lane & 15;                  // M (for A) / N (for B,C)
  int col0 = wave * 16;

  const float* a_base = A + (size_t)(row0 + l) * 64;
  v8f acc = {};
#pragma unroll
  for (int kb = 0; kb < 64; kb += 4) {
    int k0 = kb + 2 * hf;
    v2f a, b;
    a.x = a_base[k0 + 0];
    a.y = a_base[k0 + 1];
    b.x = W[(size_t)(k0 + 0) * 64 + col0 + l];
    b.y = W[(size_t)(k0 + 1) * 64 + col0 + l];
    acc = __builtin_amdgcn_wmma_f32_16x16x4_f32(false, a, false, b,
                                                (short)0, acc, false, false);
  }
  float bv = bias[col0 + l];
#pragma unroll
  for (int r = 0; r < 8; ++r) {
    int rowm = row0 + r + hf * 8;        // C/D layout: VGPR r -> M=r (lanes<16) / M=r+8
    float v = acc[r] + bv;
    out[(size_t)rowm * 64 + col0 + l] = fmaxf(v, 0.0f);
  }
}

// ---------------- layer 2 GEMM + fused log_softmax, fp32 WMMA ----------------
// one wave per 16-row strip; 16 output features live in one 16-lane half-group

__global__ __launch_bounds__(32) void k_gemm16_logsoftmax_wmma(
    const float* __restrict__ A,      // [N,64]
    const float* __restrict__ W,      // [64,16] row-major
    const float* __restrict__ bias,   // [16]
    float* __restrict__ out,          // [N,16]
    int N) {
  int row0 = blockIdx.x * 16;
  if (row0 >= N) return;
  int lane = threadIdx.x & 31;
  int hf   = lane >> 4;
  int l    = lane & 15;

  const float* a_base = A + (size_t)(row0 + l) * 64;
  v8f acc = {};
#pragma unroll
  for (int kb = 0; kb < 64; kb += 4) {
    int k0 = kb + 2 * hf;
    v2f a, b;
    a.x = a_base[k0 + 0];
    a.y = a_base[k0 + 1];
    b.x = W[(size_t)(k0 + 0) * 16 + l];
    b.y = W[(size_t)(k0 + 1) * 16 + l];
    acc = __builtin_amdgcn_wmma_f32_16x16x4_f32(false, a, false, b,
                                                (short)0, acc, false, false);
  }
  float bv = bias[l];
#pragma unroll
  for (int r = 0; r < 8; ++r) {
    float v = acc[r] + bv;
    // max over the 16 features (lanes within this half-group)
    float m = v;
#pragma unroll
    for (int off = 8; off >= 1; off >>= 1)
      m = fmaxf(m, __shfl_xor(m, off, 16));
    float e = expf(v - m);
    float s = e;
#pragma unroll
    for (int off = 8; off >= 1; off >>= 1)
      s += __shfl_xor(s, off, 16);
    int rowm = row0 + r + hf * 8;
    out[(size_t)rowm * 16 + l] = v - m - logf(s);
  }
}

// ---------------- host launch ----------------

extern "C" void kernel_launch(void* const* d_in, const int* in_sizes, int n_in,
                              void* d_out, int out_size, void* d_ws, size_t ws_size,
                              hipStream_t stream) {
  const float* x  = (const float*)d_in[0];
  const int*   ei = (const int*)d_in[1];
  const float* W1 = (const float*)d_in[2];
  const float* b1 = (const float*)d_in[3];
  const float* W2 = (const float*)d_in[4];
  const float* b2 = (const float*)d_in[5];

  const int N = in_sizes[0] / F_IN;
  const int E = in_sizes[1] / 2;
  const int* row = ei;
  const int* col = ei + E;

  auto align256 = [](size_t v) { return (v + 255) & ~(size_t)255; };
  char* ws = (char*)d_ws;
  size_t o = 0;
  float* dis  = (float*)(ws + o); o += align256((size_t)N * sizeof(float));
  float* aggr = (float*)(ws + o); o += align256((size_t)N * F_IN * sizeof(float));
  float* h    = (float*)(ws + o); o += align256((size_t)N * F_HID * sizeof(float));
  float* out  = (float*)d_out;

  const int T = 256;
  const long long ntF = (long long)N * 64;       // aggr-init thread count
  const long long ntS = (long long)E * 16;       // scatter thread count
  unsigned gN  = (unsigned)((N + T - 1) / T);
  unsigned gE  = (unsigned)((E + T - 1) / T);
  unsigned gF  = (unsigned)((ntF + T - 1) / T);
  unsigned gS  = (unsigned)((ntS + T - 1) / T);
  unsigned gMM = (unsigned)((N + 15) / 16);

  // degrees -> deg^-1/2 (in place)
  k_init_deg<<<gN, T, 0, stream>>>(dis, N);
  k_count_deg<<<gE, T, 0, stream>>>(row, dis, E);
  k_rsqrt_inplace<<<gN, T, 0, stream>>>(dis, N);

  // layer 1: aggregate x, then h = relu(aggr @ W1 + b1)
  k_init_aggr<<<gF, T, 0, stream>>>(x, dis, aggr, ntF);
  k_scatter<<<gS, T, 0, stream>>>(x, row, col, dis, aggr, E);
  k_gemm64_relu_wmma<<<gMM, 128, 0, stream>>>(aggr, W1, b1, h, N);

  // layer 2: aggregate h (reuse aggr buffer), then log_softmax(aggr @ W2 + b2)
  k_init_aggr<<<gF, T, 0, stream>>>(h, dis, aggr, ntF);
  k_scatter<<<gS, T, 0, stream>>>(h, row, col, dis, aggr, E);
  k_gemm16_logsoftmax_wmma<<<gMM, 32, 0, stream>>>(aggr, W2, b2, out, N);
}